// MusicalConstraintLoss_64381559767364
// MI455X (gfx1250) — compile-verified
//
#include <hip/hip_runtime.h>
#include <stdint.h>

// ---------------------------------------------------------------------------
// MusicalConstraintLoss on MI455X (gfx1250, wave32)
//
// Memory-bound reduction (33.5 MB read once -> ~1.4 us at 23.3 TB/s).
//   - 1 workgroup per row (512 blocks x 256 threads = 8 wave32/WG).
//   - 16 KB tiles streamed global->LDS with CDNA5 async DMA
//     (global_load_async_to_lds_b128 / ASYNCcnt), double buffered so the DMA
//     of tile t+1 overlaps the scan of tile t.
//   - Each thread scans a 16-token chunk from LDS into a segmented-scan
//     monoid state; ordered combine done intra-wave with __shfl_down
//     (wave32, no barriers), then 8 wave states folded by thread 0.
//     Only 2 block barriers per tile.
//   - Exact int64 per-row partials -> d_ws; tiny finalize kernel computes
//     the 4 f32 outputs in double precision.
// ---------------------------------------------------------------------------

#define B_ROWS  512
#define S_LEN   16384
#define THREADS 256
#define WAVES   (THREADS / 32)       // 8
#define TILE    4096                 // tokens per tile (16 KB)
#define NTILES  (S_LEN / TILE)       // 4
#define CHUNK   (TILE / THREADS)     // 16 tokens per thread per tile
#define NLD     ((TILE * 4) / (THREADS * 16))  // async b128 loads/thread/tile = 4

// Monoid state for one contiguous span of a row.
struct St {
  int t_has, t_first, t_last, t_n, t_sd;   // time-shift gap stats (positions)
  long long t_sd2;
  int v_has, v_ft, v_lt, v_cnt;            // voice-leading (note token values)
  int h_cnt;                               // harmony pairs starting in span
};

__device__ __forceinline__ St comb(const St& a, const St& b) {
  St r;
  r.t_n   = a.t_n + b.t_n;
  r.t_sd  = a.t_sd + b.t_sd;
  r.t_sd2 = a.t_sd2 + b.t_sd2;
  if (a.t_has && b.t_has) {
    int d = b.t_first - a.t_last;
    r.t_n++; r.t_sd += d; r.t_sd2 += (long long)d * (long long)d;
  }
  r.t_has   = a.t_has | b.t_has;
  r.t_first = a.t_has ? a.t_first : b.t_first;
  r.t_last  = b.t_has ? b.t_last  : a.t_last;

  r.v_cnt = a.v_cnt + b.v_cnt;
  if (a.v_has && b.v_has) {
    int iv = b.v_ft - a.v_lt;
    if (iv > 12 || iv < -12) r.v_cnt++;
  }
  r.v_has = a.v_has | b.v_has;
  r.v_ft  = a.v_has ? a.v_ft : b.v_ft;
  r.v_lt  = b.v_has ? b.v_lt : a.v_lt;

  r.h_cnt = a.h_cnt + b.h_cnt;
  return r;
}

__device__ __forceinline__ St shfl_down_st(const St& x, int s) {
  St r;
  r.t_has   = __shfl_down(x.t_has,   s);
  r.t_first = __shfl_down(x.t_first, s);
  r.t_last  = __shfl_down(x.t_last,  s);
  r.t_n     = __shfl_down(x.t_n,     s);
  r.t_sd    = __shfl_down(x.t_sd,    s);
  r.t_sd2   = __shfl_down(x.t_sd2,   s);
  r.v_has   = __shfl_down(x.v_has,   s);
  r.v_ft    = __shfl_down(x.v_ft,    s);
  r.v_lt    = __shfl_down(x.v_lt,    s);
  r.v_cnt   = __shfl_down(x.v_cnt,   s);
  r.h_cnt   = __shfl_down(x.h_cnt,   s);
  return r;
}

// Generic shared pointer -> raw LDS byte offset (low 32 bits of the shared
// aperture address are the LDS address per CDNA5 aperture rules).
__device__ __forceinline__ uint32_t lds_off_u32(const void* p) {
  return (uint32_t)(uintptr_t)p;
}

// CDNA5 async DMA: global -> LDS, 16 bytes per lane, tracked by ASYNCcnt.
__device__ __forceinline__ void async_ld_b128(uint32_t lds_off, const void* gptr) {
  asm volatile("global_load_async_to_lds_b128 %0, %1, off"
               :: "v"(lds_off), "v"(gptr)
               : "memory");
}
__device__ __forceinline__ void wait_async0() {
  asm volatile("s_wait_asynccnt 0" ::: "memory");
}

__global__ __launch_bounds__(THREADS)
void music_row_scan_kernel(const int* __restrict__ tokens,
                           long long* __restrict__ acc) {
  __shared__ int sh[2 * TILE];     // double-buffered tiles, 32 KB
  __shared__ St  swaves[WAVES];    // one state per wave (~450 B)

  const int row  = blockIdx.x;
  const int tid  = threadIdx.x;
  const int lane = tid & 31;
  const int wid  = tid >> 5;
  const int* g_row = tokens + (size_t)row * S_LEN;

  // Stage tile t into buffer buf: lanes cover the tile with contiguous 16B
  // pieces (512B contiguous per wave per issue).
  auto stage = [&](int t, int buf) {
    const char* gbase = (const char*)(g_row + t * TILE);
    uint32_t lds_base = lds_off_u32(&sh[buf * TILE]);
    #pragma unroll
    for (int k = 0; k < NLD; ++k) {
      uint32_t off = (uint32_t)((k * THREADS + tid) * 16);
      async_ld_b128(lds_base + off, gbase + off);
    }
  };

  stage(0, 0);

  St racc{};  // running row state (thread 0 only)

  for (int t = 0; t < NTILES; ++t) {
    wait_async0();       // this wave's DMA writes for tile t are in LDS
    __syncthreads();     // -> every wave's writes are visible to all
    if (t + 1 < NTILES) stage(t + 1, (t + 1) & 1);  // overlap DMA with scan

    const int* buf  = &sh[(t & 1) * TILE];
    const int  base = t * TILE + tid * CHUNK;   // absolute chunk start pos

    // This thread's 16-token chunk from LDS (4x ds_load_b128).
    int c[CHUNK];
    const int4* b4 = (const int4*)(buf + tid * CHUNK);
    #pragma unroll
    for (int q = 0; q < CHUNK / 4; ++q) {
      int4 v4 = b4[q];
      c[4 * q + 0] = v4.x; c[4 * q + 1] = v4.y;
      c[4 * q + 2] = v4.z; c[4 * q + 3] = v4.w;
    }

    // Token following this chunk (harmony pair starting at last element).
    // Sentinel past end of row fails every mask.
    int nxt;
    if (tid == THREADS - 1) {
      int a = t * TILE + TILE;
      nxt = (a < S_LEN) ? g_row[a] : 100000;
    } else {
      nxt = buf[(tid + 1) * CHUNK];
    }

    St st{};
    #pragma unroll
    for (int j = 0; j < CHUNK; ++j) {
      int tok = c[j];
      int p   = base + j;
      bool tm = (tok >= 256) && (tok < 768);  // time-shift token
      bool nm = (tok < 128);                  // note-on token (tokens >= 0)
      if (tm) {
        if (st.t_has) {
          int d = p - st.t_last;
          st.t_n++; st.t_sd += d; st.t_sd2 += (long long)d * (long long)d;
        } else { st.t_first = p; st.t_has = 1; }
        st.t_last = p;
      }
      if (nm) {
        if (st.v_has) {
          int iv = tok - st.v_lt;
          if (iv > 12 || iv < -12) st.v_cnt++;
        } else { st.v_ft = tok; st.v_has = 1; }
        st.v_lt = tok;
      }
      int tok2 = (j + 1 < CHUNK) ? c[j + 1] : nxt;
      if (nm && tok2 < 128) {
        int d12 = (tok % 12) - (tok2 % 12);
        if (d12 == 6 || d12 == -6) st.h_cnt++;
      }
    }

    // Ordered intra-wave combine (wave32 shuffles, no barriers). At level s
    // only lanes aligned to 2s fold in lane +s, which was an aligned writer
    // at the previous level, so chunk order is preserved.
    #pragma unroll
    for (int s = 1; s < 32; s <<= 1) {
      St other = shfl_down_st(st, s);
      St m = comb(st, other);
      if ((lane & (2 * s - 1)) == 0) st = m;
    }
    if (lane == 0) swaves[wid] = st;
    __syncthreads();

    // Thread 0 folds the 8 wave states in order, then into the row state.
    if (tid == 0) {
      St tacc = swaves[0];
      #pragma unroll
      for (int w = 1; w < WAVES; ++w) tacc = comb(tacc, swaves[w]);
      racc = (t == 0) ? tacc : comb(racc, tacc);
    }
    // No extra barrier needed: swaves is only rewritten after the next
    // tile's top __syncthreads(), which thread 0 reaches after these reads;
    // the tile buffer being restaged next is protected the same way.
  }

  if (tid == 0) {
    long long* a = acc + (size_t)row * 5;
    a[0] = (long long)racc.t_n;
    a[1] = (long long)racc.t_sd;
    a[2] = racc.t_sd2;
    a[3] = (long long)racc.v_cnt;
    a[4] = (long long)racc.h_cnt;
  }
}

__global__ __launch_bounds__(512)
void music_finalize_kernel(const long long* __restrict__ acc,
                           float* __restrict__ out) {
  __shared__ long long s0[512], s1[512], s2[512], s3[512], s4[512];
  const int tid = threadIdx.x;
  const long long* a = acc + (size_t)tid * 5;
  s0[tid] = a[0]; s1[tid] = a[1]; s2[tid] = a[2]; s3[tid] = a[3]; s4[tid] = a[4];
  __syncthreads();
  for (int s = 256; s > 0; s >>= 1) {
    if (tid < s) {
      s0[tid] += s0[tid + s]; s1[tid] += s1[tid + s]; s2[tid] += s2[tid + s];
      s3[tid] += s3[tid + s]; s4[tid] += s4[tid + s];
    }
    __syncthreads();
  }
  if (tid == 0) {
    double n   = (double)s0[0];
    double sd  = (double)s1[0];
    double sd2 = (double)s2[0];
    double v   = (double)s3[0];
    double h   = (double)s4[0];
    double mean = sd / fmax(n, 1.0);
    double ss   = sd2 - 2.0 * mean * sd + n * mean * mean;  // sum w*(d-mean)^2
    double var  = (n > 1.0) ? ss / (n - 1.0) : 0.0;
    double rhythm  = 0.01 * var;
    double harmony = 0.1 * h / (double)((long long)B_ROWS * (long long)S_LEN);
    double voice   = v / (double)B_ROWS;
    out[0] = (float)rhythm;
    out[1] = (float)harmony;
    out[2] = (float)voice;
    out[3] = (float)(rhythm + harmony + 0.5 * voice);
  }
}

extern "C" void kernel_launch(void* const* d_in, const int* in_sizes, int n_in,
                              void* d_out, int out_size, void* d_ws, size_t ws_size,
                              hipStream_t stream) {
  (void)in_sizes; (void)n_in; (void)out_size; (void)ws_size;
  const int* tokens = (const int*)d_in[0];
  long long* acc    = (long long*)d_ws;       // 512 rows x 5 int64 = 20 KB
  float*     out    = (float*)d_out;

  music_row_scan_kernel<<<B_ROWS, THREADS, 0, stream>>>(tokens, acc);
  music_finalize_kernel<<<1, 512, 0, stream>>>(acc, out);
}